// BesselEdgeLengthEncoding_59811714564428
// MI455X (gfx1250) — compile-verified
//
#include <hip/hip_runtime.h>
#include <math.h>

typedef float v2f __attribute__((ext_vector_type(2)));
typedef float v8f __attribute__((ext_vector_type(8)));

#define DEG        32
#define NB         8
#define WAVES_PB   8                  // 8 wave32 per 256-thread block, one node per wave
#define HC         6.0f
#define SIG_EXP    10.0f
#define W_MEAN     12.0f
#define INV_W_STD  0.25f              // 1/4
#define EPS        1e-6f
// log(10000) - log(4) - 0.5*log(2*pi)
#define LOG_C      6.9051074776516225f
#define PI_F       3.14159265358979323846f

__device__ __forceinline__ float fast_rcp(float x) {
    return __builtin_amdgcn_rcpf(x);           // v_rcp_f32, ~1 ulp
}

__device__ __forceinline__ float sigm(float x) {
    // 1/(1+exp(-x)) via v_exp_f32 + v_rcp_f32 (branchless, no IEEE div sequence)
    return fast_rcp(1.0f + __expf(-x));
}

__device__ __forceinline__ float poly50(float u) {
    // 1 - 1326 u^50 + 2600 u^51 - 1275 u^52, clipped at 0  (p = 50)
    float u2 = u * u;
    float u4 = u2 * u2;
    float u8 = u4 * u4;
    float u16 = u8 * u8;
    float u32v = u16 * u16;
    float u50 = u32v * u16 * u2;
    float u51 = u50 * u;
    float u52 = u51 * u;
    float r = 1.0f - 1326.0f * u50 + 2600.0f * u51 - 1275.0f * u52;
    return r > 0.0f ? r : 0.0f;
}

__device__ __forceinline__ float poly6(float u) {
    // 1 - 28 u^6 + 48 u^7 - 21 u^8, clipped at 0  (p = 6)
    float u2 = u * u;
    float u4 = u2 * u2;
    float u6 = u4 * u2;
    float u7 = u6 * u;
    float u8 = u7 * u;
    float r = 1.0f - 28.0f * u6 + 48.0f * u7 - 21.0f * u8;
    return r > 0.0f ? r : 0.0f;
}

__global__ __launch_bounds__(256)
void BesselEdgeFused_kernel(const float* __restrict__ norm_length,
                            const float* __restrict__ edge_length,
                            const float* __restrict__ bw,
                            float* __restrict__ out_emb,
                            float* __restrict__ out_cut,
                            float* __restrict__ out_keep,
                            int num_nodes)
{
    __shared__ float sd[WAVES_PB][DEG];
    __shared__ float senv[WAVES_PB][DEG];
    __shared__ float srank[WAVES_PB][DEG];

    const int w    = threadIdx.x >> 5;        // wave in block
    const int lane = threadIdx.x & 31;
    int node = blockIdx.x * WAVES_PB + w;
    if (node >= num_nodes) node = num_nodes - 1;   // clamp (no divergence; dup writes identical)
    const int e = node * DEG + lane;               // this lane's edge

    // ---- per-edge distance + poly50 envelope -------------------------------
    const float d   = edge_length[e];
    const float env = poly50(d * (1.0f / HC));
    sd[w][lane]   = d;
    senv[w][lane] = env;
    __syncthreads();

    // ---- ranks via V_WMMA_F32_16X16X4_F32, B = all-ones --------------------
    // A (16x4 f32) layout: lanes 0-15 -> M=lane, K={0,1}; lanes 16-31 -> M=lane-16, K={2,3}
    const int m    = lane & 15;
    const int half = lane >> 4;       // 0: K pair {0,1}, 1: K pair {2,3}
    const int i0   = m;               // row in block 0  (rows 0..15)
    const int i1   = 16 + m;          // row in block 1  (rows 16..31)
    const float di0 = sd[w][i0];
    const float di1 = sd[w][i1];

    v2f bones; bones.x = 1.0f; bones.y = 1.0f;
    v8f acc0 = {0.f,0.f,0.f,0.f,0.f,0.f,0.f,0.f};
    v8f acc1 = {0.f,0.f,0.f,0.f,0.f,0.f,0.f,0.f};

#pragma unroll
    for (int kc = 0; kc < 8; ++kc) {
        const int c0 = kc * 4 + half * 2;
        const int c1 = c0 + 1;
        const float dj0 = sd[w][c0],  dj1 = sd[w][c1];
        const float ej0 = senv[w][c0], ej1 = senv[w][c1];
        // branchless diagonal masks (keep EXEC full, avoid saveexec regions)
        const float m00 = (i0 == c0) ? 0.0f : 1.0f;
        const float m01 = (i0 == c1) ? 0.0f : 1.0f;
        const float m10 = (i1 == c0) ? 0.0f : 1.0f;
        const float m11 = (i1 == c1) ? 0.0f : 1.0f;
        v2f a0, a1;
        a0.x = m00 * ej0 * sigm(SIG_EXP * (di0 - dj0));
        a0.y = m01 * ej1 * sigm(SIG_EXP * (di0 - dj1));
        a1.x = m10 * ej0 * sigm(SIG_EXP * (di1 - dj0));
        a1.y = m11 * ej1 * sigm(SIG_EXP * (di1 - dj1));
        acc0 = __builtin_amdgcn_wmma_f32_16x16x4_f32(false, a0, false, bones,
                                                     (short)0, acc0, false, false);
        acc1 = __builtin_amdgcn_wmma_f32_16x16x4_f32(false, a1, false, bones,
                                                     (short)0, acc1, false, false);
    }

    // D (16x16 f32) layout: lane 0 VGPR v = row v (col 0); lane 16 VGPR v = row v+8.
    // All columns of D are identical (B = ones), so column 0 carries the row sums.
    if (lane == 0) {
#pragma unroll
        for (int v = 0; v < 8; ++v) { srank[w][v]      = acc0[v]; srank[w][16 + v] = acc1[v]; }
    } else if (lane == 16) {
#pragma unroll
        for (int v = 0; v < 8; ++v) { srank[w][8 + v]  = acc0[v]; srank[w][24 + v] = acc1[v]; }
    }
    __syncthreads();

    // ---- rank weights -> per-node dynamic cutoff ---------------------------
    const float rank = srank[w][lane];
    const float t    = (rank - W_MEAN) * INV_W_STD;
    float rw = __expf(LOG_C - 0.5f * t * t) + EPS;
    rw *= env;

    float dsum = rw * d;
    float wsum = rw;
#pragma unroll
    for (int off = 16; off > 0; off >>= 1) {
        dsum += __shfl_xor(dsum, off);
        wsum += __shfl_xor(wsum, off);
    }
    const float cutoff = (dsum + HC * EPS) * fast_rcp(wsum + EPS);

    // ---- per-edge outputs --------------------------------------------------
    const float keepf = (d < cutoff) ? 1.0f : 0.0f;
    const float scale = poly6(d * fast_rcp(cutoff)) * keepf;
    const float x     = norm_length[e];

    float emb[NB];
#pragma unroll
    for (int k = 0; k < NB; ++k) {
        const float wk  = bw[k];
        const float arg = PI_F * x * wk;            // sinc(x*wk) = sin(pi*x*wk)/(pi*x*wk)
        const float b   = __sinf(arg) * fast_rcp(arg);  // arg >= ~0.026, never 0
        emb[k] = b * wk * scale;
    }

    float4* o = reinterpret_cast<float4*>(out_emb + (size_t)e * NB);
    o[0] = make_float4(emb[0], emb[1], emb[2], emb[3]);
    o[1] = make_float4(emb[4], emb[5], emb[6], emb[7]);
    out_cut[e]  = cutoff;
    out_keep[e] = keepf;
}

extern "C" void kernel_launch(void* const* d_in, const int* in_sizes, int n_in,
                              void* d_out, int out_size, void* d_ws, size_t ws_size,
                              hipStream_t stream)
{
    // inputs: 0 norm_length[E], 1 edge_length[E], 2 bessel_weights[NB],
    //         3 edge_src[E] (unused), 4 edge_dst[E] (implicit: repeat(arange(N), DEG))
    const float* norm_length = (const float*)d_in[0];
    const float* edge_length = (const float*)d_in[1];
    const float* bw          = (const float*)d_in[2];

    const int E         = in_sizes[1];
    const int num_nodes = E / DEG;

    float* out_emb  = (float*)d_out;                   // [E, NB]
    float* out_cut  = out_emb + (size_t)E * NB;        // [E]
    float* out_keep = out_cut + E;                     // [E] (bool as 0.0/1.0)

    const int blocks = (num_nodes + WAVES_PB - 1) / WAVES_PB;
    BesselEdgeFused_kernel<<<blocks, WAVES_PB * 32, 0, stream>>>(
        norm_length, edge_length, bw, out_emb, out_cut, out_keep, num_nodes);
}